// SSDTableBatchedEmbeddingBags_80058190397553
// MI455X (gfx1250) — compile-verified
//
#include <hip/hip_runtime.h>
#include <hip/hip_bf16.h>
#include <stdint.h>

#if __has_include(<hip/amd_detail/amd_gfx1250_TDM.h>)
#include <hip/amd_detail/amd_gfx1250_TDM.h>
#define HAVE_TDM_HEADER 1
#endif

// Problem constants (from the reference).
#define NUM_TABLES 8
#define BATCH      8192      // 2^13
#define BAG_L      20
#define DIM        128       // 512 bytes per row
#define WAVES_PER_BLOCK 8    // 256 threads, wave32

typedef float    v4f   __attribute__((ext_vector_type(4)));
typedef uint32_t u32x4 __attribute__((ext_vector_type(4)));
typedef int32_t  i32x4 __attribute__((ext_vector_type(4)));
typedef int32_t  i32x8 __attribute__((ext_vector_type(8)));
// Exact parameter type reported by clang for the async-LDS builtin:
typedef int v4i_vs __attribute__((vector_size(16)));

// ---------------------------------------------------------------------------
// Primary (launched) kernel: one wave32 per bag.
//   lane i handles float4 chunk i of the 512B row.
//   20 global_load_b128 in flight per wave; indices fetched via SMEM (scalar).
//   Output stored non-temporally (written once, never re-read).
// ---------------------------------------------------------------------------
__global__ __launch_bounds__(256)
void tbe_fwd_direct(const int* __restrict__ indices,   // [T,B,L] int32
                    const float* __restrict__ weights, // [T*ROWS, D]
                    const int* __restrict__ cumsum,    // [T+1]
                    float* __restrict__ out)           // [B, T*D]
{
    const int lane = threadIdx.x & 31;
    const int wave = threadIdx.x >> 5;
    int bag = blockIdx.x * WAVES_PER_BLOCK + wave;     // bag = t*BATCH + b
    // bag is wave-uniform; make it provably scalar so index loads become s_load.
    bag = __builtin_amdgcn_readfirstlane(bag);

    const int t = bag >> 13;           // / BATCH
    const int b = bag & (BATCH - 1);

    const int64_t rowOff = (int64_t)cumsum[t];                 // scalar load
    const int* __restrict__ bagIdx = indices + (size_t)bag * BAG_L;

    v4f acc = {0.f, 0.f, 0.f, 0.f};
#pragma unroll
    for (int l = 0; l < BAG_L; ++l) {
        const int64_t row = rowOff + (int64_t)bagIdx[l];       // scalar idx
        const v4f* src = (const v4f*)(weights + (size_t)row * DIM) + lane;
        acc += *src;                                           // global_load_b128
    }

    v4f* dst = (v4f*)(out + (size_t)b * (NUM_TABLES * DIM) + (size_t)t * DIM) + lane;
    __builtin_nontemporal_store(acc, dst);                     // NT store hint
}

// ---------------------------------------------------------------------------
// CDNA5 async-to-LDS staging variant (compiled, NOT launched yet — promoted
// once the emitted codegen is confirmed in the disasm). One wave per block.
// ASYNCcnt tracks the 20 outstanding global->LDS DMAs; no VGPR destinations,
// so occupancy is LDS-limited only (10KB/wave -> 32 waves per 320KB WGP).
// ---------------------------------------------------------------------------
#if defined(__AMDGCN__)
__global__ __launch_bounds__(32)
void tbe_fwd_async_lds(const int* __restrict__ indices,
                       const float* __restrict__ weights,
                       const int* __restrict__ cumsum,
                       float* __restrict__ out)
{
    __shared__ float stage[BAG_L * DIM];   // 10 KB: 20 staged rows
    const int lane = threadIdx.x & 31;
    int bag = __builtin_amdgcn_readfirstlane((int)blockIdx.x);
    const int t = bag >> 13;
    const int b = bag & (BATCH - 1);
    const int64_t rowOff = (int64_t)cumsum[t];
    const int* bagIdx = indices + (size_t)bag * BAG_L;

#if __has_builtin(__builtin_amdgcn_global_load_async_to_lds_b128)
#pragma unroll
    for (int l = 0; l < BAG_L; ++l) {
        const int64_t row = rowOff + (int64_t)bagIdx[l];
        const void* src = (const void*)((const v4f*)(weights + (size_t)row * DIM) + lane);
        void* dst = (void*)(&stage[l * DIM + lane * 4]);
        __builtin_amdgcn_global_load_async_to_lds_b128(
            (__attribute__((address_space(1))) v4i_vs*)(uintptr_t)src,
            (__attribute__((address_space(3))) v4i_vs*)(uint32_t)(uintptr_t)dst,
            0, 0);
    }
#if __has_builtin(__builtin_amdgcn_s_wait_asynccnt)
    __builtin_amdgcn_s_wait_asynccnt(0);                       // LDS now written
#endif
#else
    // Fallback staging so the kernel is self-consistent even without the builtin.
#pragma unroll
    for (int l = 0; l < BAG_L; ++l) {
        const int64_t row = rowOff + (int64_t)bagIdx[l];
        ((v4f*)&stage[l * DIM])[lane] = ((const v4f*)(weights + (size_t)row * DIM))[lane];
    }
#endif

    v4f acc = {0.f, 0.f, 0.f, 0.f};
#pragma unroll
    for (int l = 0; l < BAG_L; ++l)
        acc += ((const v4f*)&stage[l * DIM])[lane];            // ds_load_b128

    v4f* dst = (v4f*)(out + (size_t)b * (NUM_TABLES * DIM) + (size_t)t * DIM) + lane;
    __builtin_nontemporal_store(acc, dst);
}

// ---------------------------------------------------------------------------
// TDM gather-mode probe (compiled, NOT launched): the Tensor Data Mover's
// gather mode is exactly an embedding-row gather (8 rows / op, 32-bit
// indices, data_size=4B, tile_dim0=128 elems). Descriptor per CDNA5 ISA.
// Toolchain is the 6-arg (clang-23) form: (g0, g1, g2, g3, g4, cpol).
// ---------------------------------------------------------------------------
__global__ __launch_bounds__(32)
void tbe_tdm_gather_probe(const int* __restrict__ indices,
                          const float* __restrict__ weights,
                          const int* __restrict__ cumsum,
                          float* __restrict__ out)
{
#if __has_builtin(__builtin_amdgcn_tensor_load_to_lds)
    __shared__ float tile[8 * DIM];    // 8 gathered rows, 4 KB
    int bag = __builtin_amdgcn_readfirstlane((int)blockIdx.x);
    const int t = bag >> 13;
    const int b = bag & (BATCH - 1);
    const int* bagIdx = indices + (size_t)bag * BAG_L;
    const uint64_t tableBase =
        (uint64_t)(uintptr_t)(weights + (size_t)cumsum[t] * DIM);
    const uint32_t ldsAddr = (uint32_t)(uintptr_t)&tile[0];

    // D# group 0: count=1, gather_mode=1, 32-bit indices, lds/global addr, type=2
    u32x4 g0;
    g0[0] = 0xC0000001u;
    g0[1] = ldsAddr;
    g0[2] = (uint32_t)tableBase;
    g0[3] = (uint32_t)((tableBase >> 32) & 0x1FFFFFFu) | (2u << 30);

    // D# group 1: data_size=4B(code 2); tensor_dim0=128; tensor_dim1=rows;
    // tile_dim0=128; tile_dim1=8 valid indices; tensor_dim0_stride=128.
    const uint32_t rowsInTable = 250000u;
    i32x8 g1;
    g1[0] = (int)(2u << 16);
    g1[1] = (int)((uint32_t)DIM << 16);
    g1[2] = (int)((rowsInTable & 0xFFFFu) << 16);
    g1[3] = (int)((rowsInTable >> 16) | ((uint32_t)DIM << 16));
    g1[4] = 8;            // tile_dim1 = #valid gather indices
    g1[5] = DIM;          // tensor_dim0_stride (data_size units)
    g1[6] = 0;
    g1[7] = 0;

    // D# groups 2/3: eight 32-bit row indices.
    i32x4 g2, g3;
    g2[0] = bagIdx[0]; g2[1] = bagIdx[1]; g2[2] = bagIdx[2]; g2[3] = bagIdx[3];
    g3[0] = bagIdx[4]; g3[1] = bagIdx[5]; g3[2] = bagIdx[6]; g3[3] = bagIdx[7];

    // 6-arg form (this toolchain): extra i32x8 group, then cpol.
    i32x8 g4 = {0, 0, 0, 0, 0, 0, 0, 0};
    __builtin_amdgcn_tensor_load_to_lds(g0, g1, g2, g3, g4, 0); // TENSORcnt++
    __builtin_amdgcn_s_wait_tensorcnt(0);

    const int lane = threadIdx.x & 31;
    v4f acc = {0.f, 0.f, 0.f, 0.f};
#pragma unroll
    for (int r = 0; r < 8; ++r)
        acc += ((const v4f*)&tile[r * DIM])[lane];
    v4f* dst = (v4f*)(out + (size_t)b * (NUM_TABLES * DIM) + (size_t)t * DIM) + lane;
    __builtin_nontemporal_store(acc, dst);
#else
    (void)indices; (void)weights; (void)cumsum; (void)out;
#endif
}
#endif // __AMDGCN__

// ---------------------------------------------------------------------------
// Host-side launch.
// ---------------------------------------------------------------------------
extern "C" void kernel_launch(void* const* d_in, const int* in_sizes, int n_in,
                              void* d_out, int out_size, void* d_ws, size_t ws_size,
                              hipStream_t stream) {
    (void)in_sizes; (void)n_in; (void)out_size; (void)d_ws; (void)ws_size;
    const int*   indices = (const int*)d_in[0];    // [T,B,L]
    const float* weights = (const float*)d_in[1];  // [T*ROWS, D]
    const int*   cumsum  = (const int*)d_in[2];    // [T+1]
    float*       out     = (float*)d_out;          // [B, T*D]

    const int total_bags = NUM_TABLES * BATCH;                 // 65536
    const dim3 grid(total_bags / WAVES_PER_BLOCK);             // 8192 blocks
    const dim3 block(32 * WAVES_PER_BLOCK);                    // 256 threads
    tbe_fwd_direct<<<grid, block, 0, stream>>>(indices, weights, cumsum, out);
}